// CoxLoss_26371099198524
// MI455X (gfx1250) — compile-verified
//
#include <hip/hip_runtime.h>
#include <stdint.h>

// Problem size: N = 8388608 = 2^23 exactly (power of two -> exact tiling).
#define NTOT      (1u << 23)
#define NB_SORT   2048          // radix blocks
#define TILE_SORT 4096          // elements per radix block (16 rounds x 256 thr)
#define NB_SCAN   4096          // scan/gather blocks
#define TILE_SCAN 2048          // elements per scan block (8 / thread)
#define HIST_N    (256 * NB_SORT)  // 524288 counters
#define HB        128           // hist-scan blocks (tile 4096)

typedef float v8f __attribute__((ext_vector_type(8)));
typedef float v2f __attribute__((ext_vector_type(2)));

// ---------------- wave32 helpers ----------------
__device__ __forceinline__ unsigned ballot32(bool p) {
#if __has_builtin(__builtin_amdgcn_ballot_w32)
  return __builtin_amdgcn_ballot_w32(p);
#else
  return (unsigned)__ballot(p);
#endif
}

__device__ __forceinline__ float wave_incl_scan_f(float x, int lane) {
#pragma unroll
  for (int d = 1; d < 32; d <<= 1) {
    float o = __shfl_up(x, d, 32);
    if (lane >= d) x += o;
  }
  return x;
}
__device__ __forceinline__ unsigned wave_incl_scan_u(unsigned x, int lane) {
#pragma unroll
  for (int d = 1; d < 32; d <<= 1) {
    unsigned o = __shfl_up(x, d, 32);
    if (lane >= d) x += o;
  }
  return x;
}
__device__ __forceinline__ float wave_red_f(float x) {
#pragma unroll
  for (int d = 16; d >= 1; d >>= 1) x += __shfl_xor(x, d, 32);
  return x;
}
__device__ __forceinline__ unsigned wave_red_u(unsigned x) {
#pragma unroll
  for (int d = 16; d >= 1; d >>= 1) x += __shfl_xor(x, d, 32);
  return x;
}

// ---------------- sort: init keys ----------------
// times >= 0 -> float bit pattern is order-preserving; ~bits => ascending sort
// by key == descending sort by time. Stable LSD + index payload reproduces
// jnp.argsort(-times) exactly (ties keep original order).
__global__ void init_kernel(const float* __restrict__ times,
                            unsigned* __restrict__ k, unsigned* __restrict__ idx) {
  unsigned i = blockIdx.x * 256u + threadIdx.x;
  k[i] = ~__float_as_uint(times[i]);
  idx[i] = i;
}

// ---------------- sort: per-block digit histogram (digit-major output) ------
__global__ void hist_kernel(const unsigned* __restrict__ kIn,
                            unsigned* __restrict__ H, int shift) {
  __shared__ unsigned h[256];
  int t = threadIdx.x, b = blockIdx.x;
  h[t] = 0;
  __syncthreads();
  long tileBase = (long)b * TILE_SORT;
#pragma unroll
  for (int r = 0; r < 16; ++r) {
    long i = tileBase + r * 256 + t;
    if (r < 15) __builtin_prefetch(&kIn[i + 256], 0, 0);  // global_prefetch_b8
    unsigned key = kIn[i];
    atomicAdd(&h[(key >> shift) & 255u], 1u);
  }
  __syncthreads();
  H[(size_t)t * NB_SORT + b] = h[t];  // digit-major: flat exclusive scan = bases
}

// ---------------- generic u32 reduce/scan for the 524288 counters -----------
__global__ void uscan_reduce(const unsigned* __restrict__ A, unsigned* __restrict__ part) {
  __shared__ unsigned wt[8];
  int t = threadIdx.x, b = blockIdx.x, lane = t & 31, wid = t >> 5;
  long base = (long)b * 4096;
  unsigned s = 0;
#pragma unroll
  for (int j = 0; j < 16; ++j) s += A[base + t + j * 256];
  s = wave_red_u(s);
  if (lane == 0) wt[wid] = s;
  __syncthreads();
  if (t == 0) {
    unsigned tot = 0;
    for (int w = 0; w < 8; ++w) tot += wt[w];
    part[b] = tot;
  }
}

__global__ void block_excl_scan_u32(unsigned* __restrict__ A, int n) {
  __shared__ unsigned wt[8];
  int t = threadIdx.x, lane = t & 31, wid = t >> 5;
  unsigned carry = 0;
  for (int base = 0; base < n; base += 256) {
    unsigned x = (base + t < n) ? A[base + t] : 0u;
    unsigned incl = wave_incl_scan_u(x, lane);
    if (lane == 31) wt[wid] = incl;
    __syncthreads();
    unsigned wbase = 0, tot = 0;
    for (int w = 0; w < wid; ++w) wbase += wt[w];
    for (int w = 0; w < 8; ++w) tot += wt[w];
    if (base + t < n) A[base + t] = carry + wbase + incl - x;
    carry += tot;
    __syncthreads();
  }
}

__global__ void uscan_scan(unsigned* __restrict__ A, const unsigned* __restrict__ part) {
  __shared__ unsigned wt[8];
  int t = threadIdx.x, b = blockIdx.x, lane = t & 31, wid = t >> 5;
  long base = (long)b * 4096 + (long)t * 16;
  unsigned v[16], ts = 0;
#pragma unroll
  for (int j = 0; j < 16; ++j) { v[j] = A[base + j]; ts += v[j]; }
  unsigned incl = wave_incl_scan_u(ts, lane);
  if (lane == 31) wt[wid] = incl;
  __syncthreads();
  unsigned wbase = 0;
  for (int w = 0; w < wid; ++w) wbase += wt[w];
  unsigned run = part[b] + wbase + incl - ts;
#pragma unroll
  for (int j = 0; j < 16; ++j) { unsigned cur = run; run += v[j]; A[base + j] = cur; }
}

// ---------------- sort: STABLE scatter via wave32 ballot multisplit ---------
// Striped rounds: round-major order == memory order => stability. Cross-wave
// ranks via LDS cnt[wave][digit] table + barriers (deterministic, no races).
__global__ void scatter_kernel(const unsigned* __restrict__ kIn,
                               const unsigned* __restrict__ iIn,
                               unsigned* __restrict__ kOut,
                               unsigned* __restrict__ iOut,
                               const unsigned* __restrict__ histS, int shift) {
  __shared__ unsigned cntW[8][256];
  __shared__ unsigned baseD[256];
  int t = threadIdx.x, b = blockIdx.x, lane = t & 31, wid = t >> 5;
  baseD[t] = histS[(size_t)t * NB_SORT + b];
#pragma unroll
  for (int w = 0; w < 8; ++w) cntW[w][t] = 0;
  __syncthreads();
  long tileBase = (long)b * TILE_SORT;
#pragma unroll 1
  for (int r = 0; r < 16; ++r) {
    long i = tileBase + r * 256 + t;
    unsigned key = kIn[i];
    unsigned idx = iIn[i];
    unsigned dig = (key >> shift) & 255u;
    unsigned mask = 0xffffffffu;
#pragma unroll
    for (int bit = 0; bit < 8; ++bit) {
      unsigned bal = ballot32((dig >> bit) & 1u);
      mask &= ((dig >> bit) & 1u) ? bal : ~bal;
    }
    unsigned lower = mask & ((1u << lane) - 1u);
    unsigned lrank = __popc(lower);
    if (lower == 0u) cntW[wid][dig] = __popc(mask);  // per-(wave,digit) leader
    __syncthreads();
    unsigned cross = 0;
    for (int w = 0; w < wid; ++w) cross += cntW[w][dig];
    unsigned dst = baseD[dig] + cross + lrank;
    kOut[dst] = key;
    iOut[dst] = idx;
    __syncthreads();
    unsigned s = 0;
#pragma unroll
    for (int w = 0; w < 8; ++w) { s += cntW[w][t]; cntW[w][t] = 0; }
    baseD[t] += s;  // running per-digit base for next round
    __syncthreads();
  }
}

// ---------------- gather exp(clip(lr)), censor packed in sign bit -----------
__global__ void gather_reduce(const unsigned* __restrict__ sidx,
                              const float* __restrict__ logr,
                              const int* __restrict__ cens,
                              float* __restrict__ ev, float* __restrict__ sv,
                              float* __restrict__ fpart) {
  __shared__ float wt[8];
  int t = threadIdx.x, b = blockIdx.x, lane = t & 31, wid = t >> 5;
  long base = (long)b * TILE_SCAN;
  float s = 0.f;
#pragma unroll
  for (int j = 0; j < 8; ++j) {
    long i = base + t + j * 256;
    unsigned id = sidx[i];
    float lr = fminf(fmaxf(logr[id], -10.f), 10.f);
    float e = expf(lr);                 // in [e^-10, e^10]: sign bit is free
    ev[i] = cens[id] ? -e : e;          // sign bit == event flag
    sv[i] = lr;
    s += e;
  }
  s = wave_red_f(s);
  if (lane == 0) wt[wid] = s;
  __syncthreads();
  if (t == 0) {
    float tot = 0.f;
    for (int w = 0; w < 8; ++w) tot += wt[w];
    fpart[b] = tot;
  }
}

__global__ void block_excl_scan_f32(float* __restrict__ A, int n) {
  __shared__ float wt[8];
  int t = threadIdx.x, lane = t & 31, wid = t >> 5;
  float carry = 0.f;
  for (int base = 0; base < n; base += 256) {
    float x = (base + t < n) ? A[base + t] : 0.f;
    float incl = wave_incl_scan_f(x, lane);
    if (lane == 31) wt[wid] = incl;
    __syncthreads();
    float wbase = 0.f, tot = 0.f;
    for (int w = 0; w < wid; ++w) wbase += wt[w];
    for (int w = 0; w < 8; ++w) tot += wt[w];
    if (base + t < n) A[base + t] = carry + wbase + incl - x;
    carry += tot;
    __syncthreads();
  }
}

// ---------------- WMMA-based inclusive scan + fused finalize ----------------
// Per wave: 256-element segment as 16x16 column-major matrix X (X[r][c] =
// x[c*16+r]). P = L * X via 4 chained V_WMMA_F32_16X16X4_F32 (L = lower-
// triangular ones, K-chunked) gives within-column inclusive scans; a 16-wide
// shuffle scan of the column totals finishes the segment. C/D layout puts 8
// consecutive results per lane. csum never touches memory (fused finalize).
__global__ void scan_finalize_kernel(const float* __restrict__ ev,
                                     const float* __restrict__ sv,
                                     const float* __restrict__ fpart,
                                     float* __restrict__ bsum,
                                     float* __restrict__ bcnt) {
  __shared__ float xs[TILE_SCAN];
  __shared__ float segLds[8];
  __shared__ float wrS[8], wrC[8];
  int t = threadIdx.x, b = blockIdx.x, lane = t & 31, wid = t >> 5;
  long base = (long)b * TILE_SCAN;
#pragma unroll
  for (int j = 0; j < 8; ++j) xs[t + j * 256] = ev[base + t + j * 256];
  __syncthreads();

  int m = lane & 15;   // column (N)
  int h = lane >> 4;   // half: row/K group
  int segL = wid * 256;
  v8f c = {0.f, 0.f, 0.f, 0.f, 0.f, 0.f, 0.f, 0.f};

#if __has_builtin(__builtin_amdgcn_wmma_f32_16x16x4_f32)
#pragma unroll
  for (int k = 0; k < 4; ++k) {
    int K0 = 4 * k + 2 * h;  // this lane's two K indices: K0, K0+1
    v2f a, bv;
    a.x = (K0 <= m) ? 1.0f : 0.0f;       // A = L[:, K-chunk] from lane id
    a.y = (K0 + 1 <= m) ? 1.0f : 0.0f;
    int xi = segL + m * 16 + K0;          // B = X[K-chunk, :] from LDS
    bv.x = __builtin_fabsf(xs[xi]);
    bv.y = __builtin_fabsf(xs[xi + 1]);
    c = __builtin_amdgcn_wmma_f32_16x16x4_f32(false, a, false, bv,
                                              (short)0, c, false, false);
  }
#else
  // Scalar fallback producing the identical C/D layout: lane holds prefix of
  // column m up to rows 8h..8h+7.
  {
    float run = 0.f;
    int rbase = segL + m * 16;
#pragma unroll
    for (int r = 0; r < 16; ++r) {
      run += __builtin_fabsf(xs[rbase + r]);
      int j = r - 8 * h;
      if (j >= 0 && j < 8) c[j] = run;
    }
  }
#endif

  // Column totals T[m] live in c[7] of upper-half lanes (row 15).
  float tt = c[7];
#pragma unroll
  for (int d = 1; d < 16; d <<= 1) {
    float o = __shfl_up(tt, d, 32);
    if (m >= d) tt += o;
  }
  float sx = __shfl_up(tt, 1, 32);
  if (m == 0) sx = 0.f;
  float scol = __shfl(sx, 16 + m, 32);      // exclusive prefix of column totals
  float segTot = __shfl(tt, 31, 32);
#pragma unroll
  for (int j = 0; j < 8; ++j) c[j] += scol; // segment-inclusive scan done

  if (lane == 0) segLds[wid] = segTot;
  __syncthreads();
  float segBase = 0.f;
  for (int w = 0; w < wid; ++w) segBase += segLds[w];  // deterministic order
  float gbase = fpart[b] + segBase;

  // Fused finalize: lane's 8 results are contiguous at local index `run`.
  int run = segL + m * 16 + 8 * h;
  long gpos = base + run;
  float lsum = 0.f, lcnt = 0.f;
#pragma unroll
  for (int j = 0; j < 8; ++j) {
    float csum = c[j] + gbase;
    unsigned sbits = __float_as_uint(xs[run + j]);
    if (sbits >> 31) {  // event (censor==1)
      lsum += sv[gpos + j] - logf(csum + 1e-15f);
      lcnt += 1.f;
    }
  }
  lsum = wave_red_f(lsum);
  lcnt = wave_red_f(lcnt);
  if (lane == 0) { wrS[wid] = lsum; wrC[wid] = lcnt; }
  __syncthreads();
  if (t == 0) {
    float S = 0.f, C = 0.f;
    for (int w = 0; w < 8; ++w) { S += wrS[w]; C += wrC[w]; }
    bsum[b] = S;
    bcnt[b] = C;
  }
}

// ---------------- final deterministic reduction ----------------
__global__ void final_kernel(const float* __restrict__ bsum,
                             const float* __restrict__ bcnt,
                             float* __restrict__ out) {
  __shared__ float wsS[8], wsC[8];
  int t = threadIdx.x, lane = t & 31, wid = t >> 5;
  float s = 0.f, n = 0.f;
  for (int i = t; i < NB_SCAN; i += 256) { s += bsum[i]; n += bcnt[i]; }
  s = wave_red_f(s);
  n = wave_red_f(n);
  if (lane == 0) { wsS[wid] = s; wsC[wid] = n; }
  __syncthreads();
  if (t == 0) {
    float S = 0.f, C = 0.f;
    for (int w = 0; w < 8; ++w) { S += wsS[w]; C += wsC[w]; }
    out[0] = -S / fmaxf(C, 1.0f);
  }
}

// ---------------- host orchestration ----------------
extern "C" void kernel_launch(void* const* d_in, const int* in_sizes, int n_in,
                              void* d_out, int out_size, void* d_ws, size_t ws_size,
                              hipStream_t stream) {
  (void)in_sizes; (void)n_in; (void)out_size; (void)ws_size;
  const float* logr  = (const float*)d_in[0];
  const float* times = (const float*)d_in[1];
  const int*   cens  = (const int*)d_in[2];
  float* out = (float*)d_out;

  char* w = (char*)d_ws;
  const size_t nb = (size_t)NTOT * 4u;
  unsigned* kA = (unsigned*)(w);
  unsigned* iA = (unsigned*)(w + nb);
  unsigned* kB = (unsigned*)(w + 2 * nb);
  unsigned* iB = (unsigned*)(w + 3 * nb);
  unsigned* H     = (unsigned*)(w + 4 * nb);
  unsigned* upart = (unsigned*)(w + 4 * nb + (size_t)HIST_N * 4u);
  float*    fpart = (float*)(w + 4 * nb + (size_t)HIST_N * 4u + 1024u);
  float*    bsum  = fpart + NB_SCAN;
  float*    bcnt  = bsum + NB_SCAN;

  init_kernel<<<NTOT / 256, 256, 0, stream>>>(times, kA, iA);

  unsigned *ki = kA, *ii = iA, *ko = kB, *io = iB;
  for (int p = 0; p < 4; ++p) {
    int shift = 8 * p;
    hist_kernel<<<NB_SORT, 256, 0, stream>>>(ki, H, shift);
    uscan_reduce<<<HB, 256, 0, stream>>>(H, upart);
    block_excl_scan_u32<<<1, 256, 0, stream>>>(upart, HB);
    uscan_scan<<<HB, 256, 0, stream>>>(H, upart);
    scatter_kernel<<<NB_SORT, 256, 0, stream>>>(ki, ii, ko, io, H, shift);
    unsigned* tk = ki; ki = ko; ko = tk;
    unsigned* ti = ii; ii = io; io = ti;
  }
  // After 4 passes sorted (key, idx) are back in (kA, iA); kB/iB are free and
  // reused as ev (exp with censor sign) and sv (clipped log-risk).
  float* ev = (float*)ko;  // == kB
  float* sv = (float*)io;  // == iB

  gather_reduce<<<NB_SCAN, 256, 0, stream>>>(ii, logr, cens, ev, sv, fpart);
  block_excl_scan_f32<<<1, 256, 0, stream>>>(fpart, NB_SCAN);
  scan_finalize_kernel<<<NB_SCAN, 256, 0, stream>>>(ev, sv, fpart, bsum, bcnt);
  final_kernel<<<1, 256, 0, stream>>>(bsum, bcnt, out);
}